// PhysicsInformedGNN_9792525435129
// MI455X (gfx1250) — compile-verified
//
#include <hip/hip_runtime.h>
#include <hip/hip_bf16.h>

// ---------------------------------------------------------------------------
// PhysicsInformedGNN forward for gfx1250 (MI455X), wave32, fp32 WMMA.
//
// mp_layer folding:  msg = x[src]@Wm_top + ea*(w_e@Wm_bot) + (b_e@Wm_bot+b_msg)
//   => segment_sum(msg) = Xagg@Wm_top + S*u + deg*c   (Xagg/S/deg via atomics)
// Dense node GEMMs use V_WMMA_F32_16X16X4_F32 (exact fp32, 16x16 tiles).
// ---------------------------------------------------------------------------

#define DEV __device__ __forceinline__

typedef __attribute__((ext_vector_type(2))) float v2f;
typedef __attribute__((ext_vector_type(8))) float v8f;

static constexpr int   kN  = 100000;            // nodes (divisible by 32)
static constexpr int   kE  = 1600000;           // edges
static constexpr int   kEN = kE + kN;           // edges + self loops
static constexpr float kEps = 1e-5f;

// ---------------- WMMA helpers (CDNA5 f32 16x16x4) --------------------------

DEV v8f wmma4(v8f acc, v2f a, v2f b) {
  // 8-arg form: (neg_a, A, neg_b, B, c_mod, C, reuse_a, reuse_b)
  return __builtin_amdgcn_wmma_f32_16x16x4_f32(
      false, a, false, b, (short)0, acc, false, false);
}

// One 16x16 output tile: rows [rowBase,rowBase+16) of A (row-major, lda=K),
// cols [colBase,colBase+16) of B (row-major K x ldb).  Optional fused
// x -> relu(x*scale[k]+shift[k]) transform on A loads (batch-norm + relu).
DEV v8f gemm_tile_f32(const float* __restrict__ A, int lda, long long rowBase,
                      const float* __restrict__ B, int ldb, int colBase,
                      int K, v8f acc, int lane,
                      const float* __restrict__ in_scale,
                      const float* __restrict__ in_shift) {
  const int m  = lane & 15;          // A row within tile / B col within tile
  const int kh = (lane >> 4) << 1;   // lanes 16-31 hold K+2,K+3
  const float* arow = A + (rowBase + m) * (long long)lda;
  for (int k = 0; k < K; k += 4) {
    float a0 = arow[k + kh];
    float a1 = arow[k + kh + 1];
    if (in_scale) {
      a0 = fmaxf(a0 * in_scale[k + kh]     + in_shift[k + kh],     0.f);
      a1 = fmaxf(a1 * in_scale[k + kh + 1] + in_shift[k + kh + 1], 0.f);
    }
    v2f a; a.x = a0; a.y = a1;
    v2f b;
    b.x = B[(long long)(k + kh)     * ldb + colBase + m];
    b.y = B[(long long)(k + kh + 1) * ldb + colBase + m];
    acc = wmma4(acc, a, b);
  }
  return acc;
}

// ---------------- small utility kernels -------------------------------------

__global__ void fill_kernel(float* __restrict__ p, float v, long long n) {
  long long i = (long long)blockIdx.x * blockDim.x + threadIdx.x;
  if (i < n) p[i] = v;
}

// u = w_edge @ Wmsg_bot ; c = b_edge @ Wmsg_bot + b_msg   for conv1 & conv2.
__global__ void prep_fold_kernel(const float* __restrict__ we1, const float* __restrict__ be1,
                                 const float* __restrict__ wm1, const float* __restrict__ bm1,
                                 const float* __restrict__ we2, const float* __restrict__ be2,
                                 const float* __restrict__ wm2, const float* __restrict__ bm2,
                                 float* __restrict__ UC) {
  int t = threadIdx.x;  // 128 threads
  if (t < 64) {
    float u = 0.f, c = 0.f;
    for (int k = 0; k < 64; ++k) {
      float wv = wm1[(16 + k) * 64 + t];   // bottom rows of W_msg1 (80x64)
      u += we1[k] * wv;  c += be1[k] * wv;
    }
    UC[t] = u;  UC[64 + t] = c + bm1[t];
  } else {
    int j = t - 64;
    float u = 0.f, c = 0.f;
    for (int k = 0; k < 64; ++k) {
      float wv = wm2[(64 + k) * 64 + j];   // bottom rows of W_msg2 (128x64)
      u += we2[k] * wv;  c += be2[k] * wv;
    }
    UC[128 + j] = u;  UC[192 + j] = c + bm2[j];
  }
}

__global__ void bn_finalize_kernel(float* __restrict__ stats,
                                   const float* __restrict__ gamma,
                                   const float* __restrict__ beta, float invN) {
  int c = threadIdx.x;  // 64 threads
  float mean = stats[c] * invN;
  float var  = stats[64 + c] * invN - mean * mean;
  float inv  = rsqrtf(var + kEps);
  float sc   = gamma[c] * inv;
  stats[128 + c] = sc;                 // scale
  stats[192 + c] = beta[c] - mean * sc;  // shift
}

// ---------------- edge scatter (conv layers) --------------------------------
// Xagg[dst] += T(X[src]); S[dst] += ea; deg[dst] += 1.  T = optional bn+relu.
__global__ void edge_scatter_kernel(const float* __restrict__ X, int FS /*log2 F*/,
                                    const int* __restrict__ src, const int* __restrict__ dst,
                                    const float* __restrict__ ea, int nedges, int nnodes,
                                    const float* __restrict__ scale, const float* __restrict__ shift,
                                    float* __restrict__ Xagg, float* __restrict__ Ssum,
                                    float* __restrict__ degv) {
  const int F = 1 << FS;
  long long i = (long long)blockIdx.x * blockDim.x + threadIdx.x;
  long long total = (long long)(nedges + nnodes) << FS;
  if (i >= total) return;
  int e = (int)(i >> FS), f = (int)(i & (F - 1));
  int s, d; float a;
  if (e < nedges) { s = src[e]; d = dst[e]; a = ea[e]; }
  else            { s = e - nedges; d = s;  a = 0.f;   }
  float v = X[(long long)s * F + f];
  if (scale) v = fmaxf(v * scale[f] + shift[f], 0.f);
  atomicAdd(&Xagg[(long long)d * F + f], v);
  if (f == 0) { atomicAdd(&Ssum[d], a); atomicAdd(&degv[d], 1.f); }
}

// ---------------- conv dense: relu(Xagg@Wa + T(Xn)@Wn + S*u + deg*c + b) ----
// Also accumulates BN statistics of the output (block LDS -> global atomics).
__global__ void conv_dense_kernel(const float* __restrict__ Xa, int Fa,
                                  const float* __restrict__ Wa,
                                  const float* __restrict__ Xn, int Fn,
                                  const float* __restrict__ Wn,
                                  const float* __restrict__ in_scale,
                                  const float* __restrict__ in_shift,
                                  const float* __restrict__ uvec,
                                  const float* __restrict__ cvec,
                                  const float* __restrict__ bnode,
                                  const float* __restrict__ Ssum,
                                  const float* __restrict__ degv,
                                  float* __restrict__ out,
                                  float* __restrict__ stats) {
  __shared__ float s_sum[64], s_sq[64];
  if (threadIdx.x < 64) { s_sum[threadIdx.x] = 0.f; s_sq[threadIdx.x] = 0.f; }
  __syncthreads();

  const int wave = threadIdx.x >> 5, lane = threadIdx.x & 31;
  const int nodeTile = blockIdx.x * 2 + (wave >> 2);  // 2 node tiles / block
  const int colTile  = wave & 3;                      // H=64 -> 4 col tiles

  v8f acc = {};
  acc = gemm_tile_f32(Xa, Fa, (long long)nodeTile * 16, Wa, 64, colTile * 16,
                      Fa, acc, lane, nullptr, nullptr);
  acc = gemm_tile_f32(Xn, Fn, (long long)nodeTile * 16, Wn, 64, colTile * 16,
                      Fn, acc, lane, in_scale, in_shift);

  const int n = lane & 15, half = lane >> 4;
  const int col = colTile * 16 + n;
  const float u = uvec[col], cc = cvec[col], bv = bnode[col];
  float lsum = 0.f, lsq = 0.f;
  for (int r = 0; r < 8; ++r) {
    long long row = (long long)nodeTile * 16 + r + half * 8;
    float v = acc[r] + Ssum[row] * u + degv[row] * cc + bv;
    v = fmaxf(v, 0.f);
    out[row * 64 + col] = v;
    lsum += v;  lsq += v * v;
  }
  atomicAdd(&s_sum[col], lsum);
  atomicAdd(&s_sq[col],  lsq);
  __syncthreads();
  if (threadIdx.x < 64) {
    atomicAdd(&stats[threadIdx.x],      s_sum[threadIdx.x]);
    atomicAdd(&stats[64 + threadIdx.x], s_sq[threadIdx.x]);
  }
}

// ---------------- GAT ------------------------------------------------------

// HG = relu(bn2(H2)) @ Wg   (64 -> 256).  blockDim=512: 16 waves = 16 col tiles.
__global__ void gat_gemm_kernel(const float* __restrict__ A, const float* __restrict__ Wg,
                                const float* __restrict__ in_scale,
                                const float* __restrict__ in_shift,
                                float* __restrict__ out) {
  const int wave = threadIdx.x >> 5, lane = threadIdx.x & 31;
  const int nodeTile = blockIdx.x, colTile = wave;
  v8f acc = {};
  acc = gemm_tile_f32(A, 64, (long long)nodeTile * 16, Wg, 256, colTile * 16,
                      64, acc, lane, in_scale, in_shift);
  const int n = lane & 15, half = lane >> 4, col = colTile * 16 + n;
  for (int r = 0; r < 8; ++r) {
    long long row = (long long)nodeTile * 16 + r + half * 8;
    out[row * 256 + col] = acc[r];
  }
}

__global__ void attn_scores_kernel(const float* __restrict__ HG,
                                   const float* __restrict__ attS,
                                   const float* __restrict__ attD,
                                   float* __restrict__ asrc, float* __restrict__ adst,
                                   int nnodes) {
  long long i = (long long)blockIdx.x * blockDim.x + threadIdx.x;
  if (i >= (long long)nnodes * 4) return;
  int nd = (int)(i >> 2), hd = (int)(i & 3);
  const float* h = HG + (long long)nd * 256 + hd * 64;
  const float* as = attS + hd * 64;
  const float* ad = attD + hd * 64;
  float ss = 0.f, sd = 0.f;
  for (int c = 0; c < 64; ++c) { float hv = h[c]; ss += hv * as[c]; sd += hv * ad[c]; }
  asrc[i] = ss;  adst[i] = sd;
}

DEV void edge_sd(int e, int nedges, const int* __restrict__ src,
                 const int* __restrict__ dst, int& s, int& d) {
  if (e < nedges) { s = src[e]; d = dst[e]; } else { s = e - nedges; d = s; }
}

DEV float leaky02(float x) { return x > 0.f ? x : 0.2f * x; }

DEV void atomic_max_float(float* addr, float val) {
  if (val >= 0.f) atomicMax((int*)addr, __float_as_int(val));
  else            atomicMin((unsigned int*)addr, __float_as_uint(val));
}

__global__ void gat_max_kernel(const int* __restrict__ src, const int* __restrict__ dst,
                               int nedges, int nnodes,
                               const float* __restrict__ asrc, const float* __restrict__ adst,
                               float* __restrict__ mmax) {
  long long i = (long long)blockIdx.x * blockDim.x + threadIdx.x;
  if (i >= (long long)(nedges + nnodes) * 4) return;
  int e = (int)(i >> 2), hd = (int)(i & 3), s, d;
  edge_sd(e, nedges, src, dst, s, d);
  float ev = leaky02(asrc[(long long)s * 4 + hd] + adst[(long long)d * 4 + hd]);
  atomic_max_float(&mmax[(long long)d * 4 + hd], ev);
}

__global__ void gat_denom_kernel(const int* __restrict__ src, const int* __restrict__ dst,
                                 int nedges, int nnodes,
                                 const float* __restrict__ asrc, const float* __restrict__ adst,
                                 const float* __restrict__ mmax, float* __restrict__ den) {
  long long i = (long long)blockIdx.x * blockDim.x + threadIdx.x;
  if (i >= (long long)(nedges + nnodes) * 4) return;
  int e = (int)(i >> 2), hd = (int)(i & 3), s, d;
  edge_sd(e, nedges, src, dst, s, d);
  float ev = leaky02(asrc[(long long)s * 4 + hd] + adst[(long long)d * 4 + hd]);
  atomicAdd(&den[(long long)d * 4 + hd], __expf(ev - mmax[(long long)d * 4 + hd]));
}

__global__ void gat_alpha_kernel(const int* __restrict__ src, const int* __restrict__ dst,
                                 int nedges, int nnodes,
                                 const float* __restrict__ asrc, const float* __restrict__ adst,
                                 const float* __restrict__ mmax, const float* __restrict__ den,
                                 float* __restrict__ alpha) {
  long long i = (long long)blockIdx.x * blockDim.x + threadIdx.x;
  if (i >= (long long)(nedges + nnodes) * 4) return;
  int e = (int)(i >> 2), hd = (int)(i & 3), s, d;
  edge_sd(e, nedges, src, dst, s, d);
  float ev = leaky02(asrc[(long long)s * 4 + hd] + adst[(long long)d * 4 + hd]);
  alpha[i] = __expf(ev - mmax[(long long)d * 4 + hd]) / den[(long long)d * 4 + hd];
}

// gout[dst][c] += 0.25 * sum_hd alpha[e][hd] * HG[src][hd*64+c]
__global__ void gat_scatter_kernel(const int* __restrict__ src, const int* __restrict__ dst,
                                   int nedges, int nnodes,
                                   const float* __restrict__ HG,
                                   const float* __restrict__ alpha,
                                   float* __restrict__ gout) {
  long long i = (long long)blockIdx.x * blockDim.x + threadIdx.x;
  if (i >= (long long)(nedges + nnodes) * 64) return;
  int e = (int)(i >> 6), c = (int)(i & 63), s, d;
  edge_sd(e, nedges, src, dst, s, d);
  const float* h  = HG + (long long)s * 256 + c;
  const float* al = alpha + (long long)e * 4;
  float acc = al[0] * h[0] + al[1] * h[64] + al[2] * h[128] + al[3] * h[192];
  atomicAdd(&gout[(long long)d * 64 + c], 0.25f * acc);
}

// y = gout + bias (in place) and accumulate BN3 statistics of y.
__global__ void gat_bias_stats_kernel(float* __restrict__ gout,
                                      const float* __restrict__ bias,
                                      float* __restrict__ stats) {
  __shared__ float s_sum[64], s_sq[64];
  int tid = threadIdx.x;
  if (tid < 64) { s_sum[tid] = 0.f; s_sq[tid] = 0.f; }
  __syncthreads();
  long long i = (long long)blockIdx.x * 256 + tid;   // grid is exact N*64/256
  int c = (int)(i & 63);
  float y = gout[i] + bias[c];
  gout[i] = y;
  atomicAdd(&s_sum[c], y);
  atomicAdd(&s_sq[c], y * y);
  __syncthreads();
  if (tid < 64) {
    atomicAdd(&stats[tid],      s_sum[tid]);
    atomicAdd(&stats[64 + tid], s_sq[tid]);
  }
}

// ---------------- MLP -------------------------------------------------------

// out = relu(T(A) @ W + bias);  T = optional bn+relu on A loads.
__global__ void mlp_gemm_kernel(const float* __restrict__ A, int K,
                                const float* __restrict__ W, int Fo,
                                const float* __restrict__ bias,
                                const float* __restrict__ in_scale,
                                const float* __restrict__ in_shift,
                                float* __restrict__ out, int nnodes) {
  int waveId = blockIdx.x * (blockDim.x >> 5) + (threadIdx.x >> 5);
  int colTiles = Fo >> 4;
  int totalTiles = (nnodes >> 4) * colTiles;
  if (waveId >= totalTiles) return;          // wave-uniform
  int nodeTile = waveId / colTiles, colTile = waveId % colTiles;
  int lane = threadIdx.x & 31;
  v8f acc = {};
  acc = gemm_tile_f32(A, K, (long long)nodeTile * 16, W, Fo, colTile * 16,
                      K, acc, lane, in_scale, in_shift);
  const int n = lane & 15, half = lane >> 4;
  const int col = colTile * 16 + n;
  const float bv = bias[col];
  for (int r = 0; r < 8; ++r) {
    long long row = (long long)nodeTile * 16 + r + half * 8;
    out[row * Fo + col] = fmaxf(acc[r] + bv, 0.f);
  }
}

__global__ void mlp3_kernel(const float* __restrict__ Z2, const float* __restrict__ W3,
                            const float* __restrict__ b3, float* __restrict__ outp,
                            int nnodes) {
  int nd = blockIdx.x * blockDim.x + threadIdx.x;
  if (nd >= nnodes) return;
  float s = b3[0];
  const float* z = Z2 + (long long)nd * 32;
  for (int k = 0; k < 32; ++k) s += z[k] * W3[k];
  outp[nd] = s;
}

// ---------------- host-side orchestration -----------------------------------

static inline int cdiv(long long a, long long b) { return (int)((a + b - 1) / b); }

extern "C" void kernel_launch(void* const* d_in, const int* in_sizes, int n_in,
                              void* d_out, int out_size, void* d_ws, size_t ws_size,
                              hipStream_t stream) {
  (void)in_sizes; (void)n_in; (void)out_size; (void)ws_size;

  // ---- inputs (setup_inputs dict order, params flattened in insertion order)
  const float* x    = (const float*)d_in[0];            // [N,16]
  const int*   eidx = (const int*)  d_in[1];            // [2,E]
  const float* ea   = (const float*)d_in[2];            // [E,1]
  const float* c1_Wn = (const float*)d_in[3];           // conv1 W_node [16,64]
  const float* c1_bn = (const float*)d_in[4];
  const float* c1_We = (const float*)d_in[5];           // [1,64]
  const float* c1_be = (const float*)d_in[6];
  const float* c1_Wm = (const float*)d_in[7];           // [80,64]
  const float* c1_bm = (const float*)d_in[8];
  const float* c2_Wn = (const float*)d_in[9];           // conv2 W_node [64,64]
  const float* c2_bn = (const float*)d_in[10];
  const float* c2_We = (const float*)d_in[11];
  const float* c2_be = (const float*)d_in[12];
  const float* c2_Wm = (const float*)d_in[13];          // [128,64]
  const float* c2_bm = (const float*)d_in[14];
  const float* g1 = (const float*)d_in[15]; const float* b1 = (const float*)d_in[16];
  const float* g2 = (const float*)d_in[17]; const float* b2 = (const float*)d_in[18];
  const float* g3 = (const float*)d_in[19]; const float* b3 = (const float*)d_in[20];
  const float* gat_W  = (const float*)d_in[21];         // [64,256]
  const float* att_s  = (const float*)d_in[22];         // [4,64]
  const float* att_d  = (const float*)d_in[23];
  const float* gat_b  = (const float*)d_in[24];         // [64]
  const float* m_W1 = (const float*)d_in[25]; const float* m_b1 = (const float*)d_in[26];
  const float* m_W2 = (const float*)d_in[27]; const float* m_b2 = (const float*)d_in[28];
  const float* m_W3 = (const float*)d_in[29]; const float* m_b3 = (const float*)d_in[30];

  const int* src = eidx;            // edge_index[0]
  const int* dst = eidx + kE;       // edge_index[1]
  float* out = (float*)d_out;

  // ---- workspace layout (floats), ~214 MB total
  float* W = (float*)d_ws;
  size_t o = 0;
  float* XAGG = W + o; o += (size_t)kN * 64;   // also reused as Z1 (mlp1 out)
  float* SSUM = W + o; o += kN;
  float* DEG  = W + o; o += kN;
  float* H1   = W + o; o += (size_t)kN * 64;   // conv1 out; reused as GOUT/y
  float* H2   = W + o; o += (size_t)kN * 64;   // conv2 out; reused as Z2
  float* HG   = W + o; o += (size_t)kN * 256;  // GAT per-head features
  float* ASRC = W + o; o += (size_t)kN * 4;
  float* ADST = W + o; o += (size_t)kN * 4;
  float* MMAX = W + o; o += (size_t)kN * 4;
  float* DEN  = W + o; o += (size_t)kN * 4;
  float* ALPH = W + o; o += (size_t)kEN * 4;
  float* UC   = W + o; o += 256;               // u1,c1,u2,c2
  float* ST1  = W + o; o += 256;               // sum,sumsq,scale,shift
  float* ST2  = W + o; o += 256;
  float* ST3  = W + o; o += 256;

  const int TB = 256;
  const float NEG_INF = -__builtin_huge_valf();

  // ---- init accumulators
  fill_kernel<<<cdiv((long long)kN * 66, TB), TB, 0, stream>>>(XAGG, 0.f, (long long)kN * 66);
  fill_kernel<<<cdiv((long long)kN * 4,  TB), TB, 0, stream>>>(MMAX, NEG_INF, (long long)kN * 4);
  fill_kernel<<<cdiv((long long)kN * 4,  TB), TB, 0, stream>>>(DEN, 0.f, (long long)kN * 4);
  fill_kernel<<<cdiv(768, TB), TB, 0, stream>>>(ST1, 0.f, 768);  // ST1..ST3 contiguous
  prep_fold_kernel<<<1, 128, 0, stream>>>(c1_We, c1_be, c1_Wm, c1_bm,
                                          c2_We, c2_be, c2_Wm, c2_bm, UC);

  // ---- conv1: scatter (F=16) + dense + bn1 stats
  edge_scatter_kernel<<<cdiv((long long)kEN * 16, TB), TB, 0, stream>>>(
      x, 4, src, dst, ea, kE, kN, nullptr, nullptr, XAGG, SSUM, DEG);
  conv_dense_kernel<<<kN / 32, TB, 0, stream>>>(
      XAGG, 16, c1_Wm,            // Xagg @ W_msg_top (rows 0..15)
      x, 16, c1_Wn,               // x @ W_node
      nullptr, nullptr,
      UC, UC + 64, c1_bn, SSUM, DEG, H1, ST1);
  bn_finalize_kernel<<<1, 64, 0, stream>>>(ST1, g1, b1, 1.f / kN);

  // ---- conv2: re-zero accum, scatter relu(bn1(H1)) (F=64), dense, bn2 stats
  fill_kernel<<<cdiv((long long)kN * 66, TB), TB, 0, stream>>>(XAGG, 0.f, (long long)kN * 66);
  edge_scatter_kernel<<<cdiv((long long)kEN * 64, TB), TB, 0, stream>>>(
      H1, 6, src, dst, ea, kE, kN, ST1 + 128, ST1 + 192, XAGG, SSUM, DEG);
  conv_dense_kernel<<<kN / 32, TB, 0, stream>>>(
      XAGG, 64, c2_Wm,            // Xagg @ W_msg2_top (rows 0..63)
      H1, 64, c2_Wn,              // relu(bn1(H1)) @ W_node2 (fused transform)
      ST1 + 128, ST1 + 192,
      UC + 128, UC + 192, c2_bn, SSUM, DEG, H2, ST2);
  bn_finalize_kernel<<<1, 64, 0, stream>>>(ST2, g2, b2, 1.f / kN);

  // ---- GAT
  gat_gemm_kernel<<<kN / 16, 512, 0, stream>>>(H2, gat_W, ST2 + 128, ST2 + 192, HG);
  attn_scores_kernel<<<cdiv((long long)kN * 4, TB), TB, 0, stream>>>(
      HG, att_s, att_d, ASRC, ADST, kN);
  gat_max_kernel<<<cdiv((long long)kEN * 4, TB), TB, 0, stream>>>(
      src, dst, kE, kN, ASRC, ADST, MMAX);
  gat_denom_kernel<<<cdiv((long long)kEN * 4, TB), TB, 0, stream>>>(
      src, dst, kE, kN, ASRC, ADST, MMAX, DEN);
  gat_alpha_kernel<<<cdiv((long long)kEN * 4, TB), TB, 0, stream>>>(
      src, dst, kE, kN, ASRC, ADST, MMAX, DEN, ALPH);
  fill_kernel<<<cdiv((long long)kN * 64, TB), TB, 0, stream>>>(H1, 0.f, (long long)kN * 64);
  gat_scatter_kernel<<<cdiv((long long)kEN * 64, TB), TB, 0, stream>>>(
      src, dst, kE, kN, HG, ALPH, H1);
  gat_bias_stats_kernel<<<(kN * 64) / TB, TB, 0, stream>>>(H1, gat_b, ST3);
  bn_finalize_kernel<<<1, 64, 0, stream>>>(ST3, g3, b3, 1.f / kN);

  // ---- MLP head: relu(bn3(y))@W1 -> relu -> @W2 -> relu -> @W3
  mlp_gemm_kernel<<<cdiv((long long)(kN / 16) * 4, 8), TB, 0, stream>>>(
      H1, 64, m_W1, 64, m_b1, ST3 + 128, ST3 + 192, XAGG, kN);   // Z1 = XAGG
  mlp_gemm_kernel<<<cdiv((long long)(kN / 16) * 2, 8), TB, 0, stream>>>(
      XAGG, 64, m_W2, 32, m_b2, nullptr, nullptr, H2, kN);       // Z2 = H2
  mlp3_kernel<<<cdiv(kN, TB), TB, 0, stream>>>(H2, m_W3, m_b3, out, kN);
}